// CorticalLayer_26336739459351
// MI455X (gfx1250) — compile-verified
//
#include <hip/hip_runtime.h>
#include <math.h>
#include <stdint.h>

// Problem sizes (B=4, S=2048 -> T=8192 tokens)
#define T_TOK 8192
#define D_IN  1024
#define D_H   512
#define N_C   16
#define N_N   256
#define MAX_TILES 528   // sum_c ceil(cnt_c/16) <= 512 + 15

typedef float v2f __attribute__((ext_vector_type(2)));
typedef float v8f __attribute__((ext_vector_type(8)));

__device__ __forceinline__ float gelu_exact(float v) {
    return 0.5f * v * (1.0f + erff(v * 0.70710678118654752440f));
}
__device__ __forceinline__ float wave_sum(float v) {
#pragma unroll
    for (int o = 16; o > 0; o >>= 1) v += __shfl_xor(v, o, 32);
    return v;
}

// ---- CDNA5 async global->LDS copy (ASYNCcnt-tracked, no VGPR round trip) ----
// LDS operand = low 32 bits of the generic pointer.
__device__ __forceinline__ void async_b128(const float* g, float* l) {
    asm volatile("global_load_async_to_lds_b128 %0, %1, off"
                 :: "v"((unsigned)(uintptr_t)l), "v"(g) : "memory");
}
template <int N>
__device__ __forceinline__ void wait_async() {
#if __has_builtin(__builtin_amdgcn_s_wait_asynccnt)
    __builtin_amdgcn_s_wait_asynccnt(N);
#else
    asm volatile("s_wait_asynccnt %0" :: "i"(N) : "memory");
#endif
}

// ---------------------------------------------------------------- kernel 0
__global__ void k0_normalize_cols(const float* __restrict__ ce, float* __restrict__ cen) {
    __shared__ float red[256];
    const int c = blockIdx.x;
    float ss = 0.f;
    for (int d = threadIdx.x; d < D_IN; d += 256) {
        float v = ce[c * D_IN + d];
        ss += v * v;
    }
    red[threadIdx.x] = ss;
    __syncthreads();
    for (int s = 128; s > 0; s >>= 1) {
        if (threadIdx.x < s) red[threadIdx.x] += red[threadIdx.x + s];
        __syncthreads();
    }
    const float inv = 1.0f / fmaxf(sqrtf(red[0]), 1e-12f);
    for (int d = threadIdx.x; d < D_IN; d += 256)
        cen[c * D_IN + d] = ce[c * D_IN + d] * inv;
}

// ---------------------------------------------------------------- kernel 1
// H[T, D_H] = gelu(X @ W1 + b1). fp32 WMMA; A+B double-buffered in LDS via
// async b128 copies (10 async instr / thread / chunk -> static wait counts).
__global__ void __launch_bounds__(128) k1_gate1(const float* __restrict__ X,
                                                const float* __restrict__ W1,
                                                const float* __restrict__ b1,
                                                float* __restrict__ H) {
    __shared__ __align__(16) float As[2][16][68];   // 272B row stride (16B mult.)
    __shared__ __align__(16) float Bs[2][64][64];
    const int tid  = threadIdx.x;
    const int wave = tid >> 5, lane = tid & 31;
    const int half = lane >> 4, l15 = lane & 15;
    const int m0 = blockIdx.y * 16;
    const int nb = blockIdx.x * 64;
    const int nl = wave * 16 + l15;

    auto issue = [&](int ib) {
        const int buf = ib & 1;
        const int kb  = ib * 64;
#pragma unroll
        for (int i = 0; i < 2; ++i) {               // A tile: 16x64 = 256 x b128
            int s = tid + i * 128;
            int row = s >> 4, col4 = (s & 15) << 2;
            async_b128(X + (size_t)(m0 + row) * D_IN + kb + col4, &As[buf][row][col4]);
        }
#pragma unroll
        for (int i = 0; i < 8; ++i) {               // B tile: 64x64 = 1024 x b128
            int s = tid + i * 128;
            int row = s >> 4, col4 = (s & 15) << 2;
            async_b128(W1 + (size_t)(kb + row) * D_H + nb + col4, &Bs[buf][row][col4]);
        }
    };

    issue(0);
    v8f acc = {};
    for (int ib = 0; ib < D_IN / 64; ++ib) {
        __syncthreads();                            // readers of buf ib^1 done
        if (ib + 1 < D_IN / 64) { issue(ib + 1); wait_async<10>(); }
        else                    { wait_async<0>(); }
        __syncthreads();                            // buf ib ready for all waves
        const int buf = ib & 1;
#pragma unroll
        for (int kk = 0; kk < 64; kk += 4) {
            const int k = kk + 2 * half;            // lanes 0-15: K0,K1; 16-31: K2,K3
            v2f a, b;
            a.x = As[buf][l15][k]; a.y = As[buf][l15][k + 1];
            b.x = Bs[buf][k][nl];  b.y = Bs[buf][k + 1][nl];
            acc = __builtin_amdgcn_wmma_f32_16x16x4_f32(false, a, false, b,
                                                        (short)0, acc, false, false);
        }
    }
    const float bias = b1[nb + nl];
#pragma unroll
    for (int v = 0; v < 8; ++v) {
        const int m = m0 + v + 8 * half;
        H[(size_t)m * D_H + nb + nl] = gelu_exact(acc[v] + bias);
    }
}

// ---------------------------------------------------------------- kernel 2
__global__ void k2_router(const float* __restrict__ X, const float* __restrict__ cen,
                          const float* __restrict__ H, const float* __restrict__ W2,
                          const float* __restrict__ b2, int* __restrict__ idx,
                          float* __restrict__ wgt, int* __restrict__ cnt) {
    const int wave = threadIdx.x >> 5;
    const int lane = threadIdx.x & 31;
    const int t = blockIdx.x * 8 + wave;
    float sim[N_C], ga[N_C];
#pragma unroll
    for (int c = 0; c < N_C; ++c) { sim[c] = 0.f; ga[c] = 0.f; }
    float xx = 0.f;
    for (int d = lane; d < D_IN; d += 32) {
        const float xv = X[(size_t)t * D_IN + d];
        xx += xv * xv;
#pragma unroll
        for (int c = 0; c < N_C; ++c) sim[c] += xv * cen[c * D_IN + d];
    }
    for (int d = lane; d < D_H; d += 32) {
        const float hv = H[(size_t)t * D_H + d];
#pragma unroll
        for (int c = 0; c < N_C; ++c) ga[c] += hv * W2[d * N_C + c];
    }
    xx = wave_sum(xx);
#pragma unroll
    for (int c = 0; c < N_C; ++c) { sim[c] = wave_sum(sim[c]); ga[c] = wave_sum(ga[c]); }
    if (lane == 0) {
        const float inv = 1.0f / fmaxf(sqrtf(xx), 1e-12f);
        float lg[N_C];
        float best = -1e30f; int bi = 0;
#pragma unroll
        for (int c = 0; c < N_C; ++c) {
            const float g = 1.0f / (1.0f + expf(-(ga[c] + b2[c])));
            const float l = sim[c] * inv + g;
            lg[c] = l;
            if (l > best) { best = l; bi = c; }   // ties -> lowest index
        }
        float se = 0.f;
#pragma unroll
        for (int c = 0; c < N_C; ++c) se += expf(lg[c] - best);
        idx[t] = bi;
        wgt[t] = 1.0f / se;
        atomicAdd(cnt + bi, 1);
    }
}

// ---------------------------------------------------------------- kernel 3
__global__ void k3_tiles(const int* __restrict__ cnt, int* __restrict__ col_start,
                         int* __restrict__ ntiles, int* __restrict__ tile_col,
                         int* __restrict__ tile_off) {
    if (threadIdx.x == 0 && blockIdx.x == 0) {
        int run = 0, tt = 0;
        for (int c = 0; c < N_C; ++c) {
            col_start[c] = run;
            const int n = cnt[c];
            for (int j = 0; j < n; j += 16) { tile_col[tt] = c; tile_off[tt] = run + j; ++tt; }
            run += n;
        }
        col_start[N_C] = run;
        ntiles[0] = tt;
    }
}

// ---------------------------------------------------------------- kernel 4
__global__ void k4_scatter(const int* __restrict__ idx, const int* __restrict__ col_start,
                           int* __restrict__ fill, int* __restrict__ perm) {
    const int t = blockIdx.x * 256 + threadIdx.x;
    const int c = idx[t];
    const int pos = col_start[c] + atomicAdd(fill + c, 1);
    perm[pos] = t;
}

// ---------------------------------------------------------------- kernel 5
// Acol[t] = gelu(x[t] @ W_cols[c(t)] + b_cols) * wgt[t]. Gathered-A WMMA with
// async double-buffered LDS staging (per-lane global+LDS addresses).
__global__ void __launch_bounds__(128) k5_colgemm(
        const float* __restrict__ X, const float* __restrict__ Wc,
        const float* __restrict__ bc, const float* __restrict__ wgt,
        const int* __restrict__ perm, const int* __restrict__ tile_col,
        const int* __restrict__ tile_off, const int* __restrict__ col_start,
        const int* __restrict__ ntiles, float* __restrict__ Acol) {
    __shared__ __align__(16) float As[2][16][68];
    __shared__ __align__(16) float Bs[2][64][64];
    __shared__ int tok_s[16];
    __shared__ float wgt_s[16];
    const int tile = blockIdx.x;
    if (tile >= ntiles[0]) return;                  // block-uniform exit
    const int c = tile_col[tile];
    const int base = tile_off[tile];
    const int cend = col_start[c + 1];
    if (threadIdx.x < 16) {
        const int p = base + threadIdx.x;
        const int tk = perm[p < cend ? p : cend - 1];   // pad with duplicate row
        tok_s[threadIdx.x] = tk;
        wgt_s[threadIdx.x] = wgt[tk];
    }
    __syncthreads();
    const int tid  = threadIdx.x;
    const int wave = tid >> 5, lane = tid & 31;
    const int half = lane >> 4, l15 = lane & 15;
    const int nb = blockIdx.y * 64;
    const int nl = wave * 16 + l15;
    const float* Bp = Wc + (size_t)c * D_IN * N_N;

    auto issue = [&](int ib) {
        const int buf = ib & 1;
        const int kb  = ib * 64;
#pragma unroll
        for (int i = 0; i < 2; ++i) {               // gathered A rows
            int s = tid + i * 128;
            int row = s >> 4, col4 = (s & 15) << 2;
            async_b128(X + (size_t)tok_s[row] * D_IN + kb + col4, &As[buf][row][col4]);
        }
#pragma unroll
        for (int i = 0; i < 8; ++i) {
            int s = tid + i * 128;
            int row = s >> 4, col4 = (s & 15) << 2;
            async_b128(Bp + (size_t)(kb + row) * N_N + nb + col4, &Bs[buf][row][col4]);
        }
    };

    issue(0);
    v8f acc = {};
    for (int ib = 0; ib < D_IN / 64; ++ib) {
        __syncthreads();
        if (ib + 1 < D_IN / 64) { issue(ib + 1); wait_async<10>(); }
        else                    { wait_async<0>(); }
        __syncthreads();
        const int buf = ib & 1;
#pragma unroll
        for (int kk = 0; kk < 64; kk += 4) {
            const int k = kk + 2 * half;
            v2f a, b;
            a.x = As[buf][l15][k]; a.y = As[buf][l15][k + 1];
            b.x = Bs[buf][k][nl];  b.y = Bs[buf][k + 1][nl];
            acc = __builtin_amdgcn_wmma_f32_16x16x4_f32(false, a, false, b,
                                                        (short)0, acc, false, false);
        }
    }
    const float bias = bc[c * N_N + nb + nl];
#pragma unroll
    for (int v = 0; v < 8; ++v) {
        const int r = v + 8 * half;
        if (base + r < cend)
            Acol[(size_t)tok_s[r] * N_N + nb + nl] =
                gelu_exact(acc[v] + bias) * wgt_s[r];
    }
}

// ---------------------------------------------------------------- kernel 6
// y = Acol @ proj_w[c*N:(c+1)*N, :] + proj_b + x ; out = LN(y)*g + b
__global__ void __launch_bounds__(128) k6_proj_ln(
        const float* __restrict__ X, const float* __restrict__ Acol,
        const float* __restrict__ Pw, const float* __restrict__ Pb,
        const float* __restrict__ lng, const float* __restrict__ lnb,
        const int* __restrict__ perm, const int* __restrict__ tile_col,
        const int* __restrict__ tile_off, const int* __restrict__ col_start,
        const int* __restrict__ ntiles, float* __restrict__ out) {
    __shared__ __align__(16) float As[16][260];     // 1040B stride (16B mult.)
    __shared__ int tok_s[16];
    extern __shared__ float Ys[];                   // 16 x 1024 floats (dynamic)
    const int tile = blockIdx.x;
    if (tile >= ntiles[0]) return;
    const int c = tile_col[tile];
    const int base = tile_off[tile];
    const int cend = col_start[c + 1];
    if (threadIdx.x < 16) {
        const int p = base + threadIdx.x;
        tok_s[threadIdx.x] = perm[p < cend ? p : cend - 1];
    }
    __syncthreads();
#pragma unroll
    for (int i = 0; i < 8; ++i) {                   // stage A: 16x256 via async
        int s = threadIdx.x + i * 128;
        int row = s >> 6, col4 = (s & 63) << 2;
        async_b128(Acol + (size_t)tok_s[row] * N_N + col4, &As[row][col4]);
    }
    wait_async<0>();
    __syncthreads();
    const int wave = threadIdx.x >> 5;
    const int lane = threadIdx.x & 31;
    const int half = lane >> 4, l15 = lane & 15;
    const float* Bp = Pw + (size_t)c * N_N * D_IN;
    for (int it = 0; it < 16; ++it) {               // 16 N-chunks of 64 -> 1024
        const int n = it * 64 + wave * 16 + l15;
        v8f acc = {};
#pragma unroll
        for (int k4 = 0; k4 < N_N; k4 += 4) {
            const int k = k4 + 2 * half;
            v2f a, b;
            a.x = As[l15][k];     a.y = As[l15][k + 1];
            b.x = Bp[(size_t)k * D_IN + n];
            b.y = Bp[(size_t)(k + 1) * D_IN + n];
            acc = __builtin_amdgcn_wmma_f32_16x16x4_f32(false, a, false, b,
                                                        (short)0, acc, false, false);
        }
        const float bias = Pb[n];
#pragma unroll
        for (int v = 0; v < 8; ++v) {
            const int r = v + 8 * half;
            Ys[r * D_IN + n] = acc[v] + bias + X[(size_t)tok_s[r] * D_IN + n];
        }
    }
    __syncthreads();
    for (int j = 0; j < 4; ++j) {                   // LN: wave w -> rows 4w..4w+3
        const int r = wave * 4 + j;
        float s = 0.f, ss = 0.f;
        for (int d = lane; d < D_IN; d += 32) {
            const float v = Ys[r * D_IN + d];
            s += v; ss += v * v;
        }
        s = wave_sum(s); ss = wave_sum(ss);
        const float mu = s * (1.0f / D_IN);
        const float var = ss * (1.0f / D_IN) - mu * mu;
        const float rs = rsqrtf(var + 1e-5f);
        if (base + r < cend) {
            const int t = tok_s[r];
            for (int d = lane; d < D_IN; d += 32)
                out[(size_t)t * D_IN + d] = (Ys[r * D_IN + d] - mu) * rs * lng[d] + lnb[d];
        }
    }
}

// ---------------------------------------------------------------- launch
extern "C" void kernel_launch(void* const* d_in, const int* in_sizes, int n_in,
                              void* d_out, int out_size, void* d_ws, size_t ws_size,
                              hipStream_t stream) {
    (void)in_sizes; (void)n_in; (void)out_size; (void)ws_size;
    const float* x   = (const float*)d_in[0];
    const float* ce  = (const float*)d_in[1];
    const float* gw1 = (const float*)d_in[2];
    const float* gb1 = (const float*)d_in[3];
    const float* gw2 = (const float*)d_in[4];
    const float* gb2 = (const float*)d_in[5];
    const float* wc  = (const float*)d_in[6];
    const float* bc  = (const float*)d_in[7];
    const float* pw  = (const float*)d_in[8];
    const float* pb  = (const float*)d_in[9];
    const float* lng = (const float*)d_in[10];
    const float* lnb = (const float*)d_in[11];
    float* out = (float*)d_out;

    // Workspace (~16.3 MB): Acol aliases H (H dead after k2).
    char* ws = (char*)d_ws;
    float* cen = (float*)(ws + 0);                  // 16x1024 f32
    int*   idx = (int*)(ws + 65536);                // 8192 i32
    float* wgt = (float*)(ws + 98304);              // 8192 f32
    char* ctrl = ws + 131072;
    int* cnt       = (int*)(ctrl + 0);
    int* fill      = (int*)(ctrl + 64);
    int* col_start = (int*)(ctrl + 128);
    int* ntiles    = (int*)(ctrl + 256);
    int* tile_col  = (int*)(ctrl + 512);
    int* tile_off  = (int*)(ctrl + 4096);
    int* perm      = (int*)(ctrl + 8192);           // 8192 i32
    float* H    = (float*)(ws + 262144);            // 8192x512 f32
    float* Acol = H;                                // 8192x256 f32, aliases H

    (void)hipMemsetAsync(ctrl, 0, 8192, stream);
    k0_normalize_cols<<<N_C, 256, 0, stream>>>(ce, cen);
    k1_gate1<<<dim3(D_H / 64, T_TOK / 16), 128, 0, stream>>>(x, gw1, gb1, H);
    k2_router<<<T_TOK / 8, 256, 0, stream>>>(x, cen, H, gw2, gb2, idx, wgt, cnt);
    k3_tiles<<<1, 32, 0, stream>>>(cnt, col_start, ntiles, tile_col, tile_off);
    k4_scatter<<<T_TOK / 256, 256, 0, stream>>>(idx, col_start, fill, perm);
    k5_colgemm<<<dim3(MAX_TILES, N_N / 64), 128, 0, stream>>>(
        x, wc, bc, wgt, perm, tile_col, tile_off, col_start, ntiles, Acol);
    k6_proj_ln<<<MAX_TILES, 128, 16 * D_IN * sizeof(float), stream>>>(
        x, Acol, pw, pb, lng, lnb, perm, tile_col, tile_off, col_start, ntiles, out);
}